// AttentionEPDGNN_7258494730680
// MI455X (gfx1250) — compile-verified
//
#include <hip/hip_runtime.h>
#include <hip/hip_bf16.h>
#include <math.h>

// ---------------------------------------------------------------------------
// Problem constants (fixed by the reference harness)
// ---------------------------------------------------------------------------
#define GN      100000     // nodes  (divisible by 16: 6250 full WMMA tiles)
#define GE      1600000    // edges
#define GDIN    32
#define GH      64
#define GDOUT   8
#define GG      64         // graphs
#define GEDGED  15

typedef __attribute__((ext_vector_type(2))) float v2f;
typedef __attribute__((ext_vector_type(8))) float v8f;

// ---------------------------------------------------------------------------
// Generic zero kernel (grid-stride)
// ---------------------------------------------------------------------------
__global__ void zero_f32(float* __restrict__ p, size_t n) {
    size_t i = (size_t)blockIdx.x * blockDim.x + threadIdx.x;
    size_t stride = (size_t)gridDim.x * blockDim.x;
    for (; i < n; i += stride) p[i] = 0.0f;
}

// ---------------------------------------------------------------------------
// Pad edge_W [2,15,64] -> [2,16,64] with zero row 15 (so WMMA K-loop is safe)
// ---------------------------------------------------------------------------
__global__ void pad_edge_w(const float* __restrict__ eW, float* __restrict__ out) {
    int idx = blockIdx.x * blockDim.x + threadIdx.x;     // 0..2047
    if (idx >= 2 * 16 * 64) return;
    int c = idx & 63;
    int k = (idx >> 6) & 15;
    int l = idx >> 10;
    out[idx] = (k < GEDGED) ? eW[(size_t)l * GEDGED * GH + (size_t)k * GH + c] : 0.0f;
}

// ---------------------------------------------------------------------------
// Encoder: h = relu(x @ enc_W + enc_b)     x:[GN,32]  enc_W:[32,64]  h:[GN,64]
// One wave computes a full 16-row x 64-col tile with V_WMMA_F32_16X16X4_F32.
// GN % 16 == 0 -> no tail handling, EXEC stays all-ones throughout.
// ---------------------------------------------------------------------------
__global__ __launch_bounds__(256) void encoder_wmma(
    const float* __restrict__ x, const float* __restrict__ W,
    const float* __restrict__ b, float* __restrict__ h)
{
    int wave = blockIdx.x * (blockDim.x >> 5) + (threadIdx.x >> 5);
    int lane = threadIdx.x & 31;
    int row0 = wave * 16;                        // always a full tile
    int m    = lane & 15;
    int half = lane >> 4;                        // 0 or 1
    int khalf = half * 2;                        // 0 or 2

    // Preload A fragments: 8 K-steps covering K=32 (8-byte vector loads)
    v2f a[8];
    const v2f* xr = (const v2f*)(x + (size_t)(row0 + m) * GDIN);
#pragma unroll
    for (int kk = 0; kk < 8; ++kk) a[kk] = xr[kk * 2 + half];

#pragma unroll
    for (int j = 0; j < 4; ++j) {                // 4 column tiles of 16
        v8f c = {};
        int col = j * 16 + m;
#pragma unroll
        for (int kk = 0; kk < 8; ++kk) {
            int k = kk * 4 + khalf;
            v2f bb;
            bb[0] = W[(size_t)k * GH + col];
            bb[1] = W[(size_t)(k + 1) * GH + col];
            c = __builtin_amdgcn_wmma_f32_16x16x4_f32(
                    false, a[kk], false, bb, (short)0, c, false, false);
        }
        float bias = b[col];
        float* hp = h + (size_t)(row0 + 8 * half) * GH + col;
#pragma unroll
        for (int rr = 0; rr < 8; ++rr) {
            float v = c[rr] + bias;
            hp[(size_t)rr * GH] = v > 0.0f ? v : 0.0f;
        }
    }
}

// ---------------------------------------------------------------------------
// Edge scatter: agg_h[d] += h[s];  agg_e[d] += edge_attr[e];  deg[d] += 1
// 64 threads per edge (2 waves); global f32 atomics. This is the HBM/atomic-
// bound hot loop (~0.9 GB traffic per layer).
// ---------------------------------------------------------------------------
__global__ __launch_bounds__(256) void scatter_edges(
    const float* __restrict__ h, const int* __restrict__ src,
    const int* __restrict__ dst, const float* __restrict__ ea,
    float* __restrict__ aggh, float* __restrict__ agge,
    float* __restrict__ deg)
{
    size_t gid = (size_t)blockIdx.x * blockDim.x + threadIdx.x;
    size_t e = gid >> 6;
    int t = (int)(gid & 63);
    if (e >= (size_t)GE) return;
    int s = src[e];
    int d = dst[e];
    atomicAdd(&aggh[(size_t)d * GH + t], h[(size_t)s * GH + t]);
    if (t < GEDGED) atomicAdd(&agge[(size_t)d * 16 + t], ea[e * GEDGED + t]);
    if (t == 0)     atomicAdd(&deg[d], 1.0f);
}

// ---------------------------------------------------------------------------
// Node update: h = relu(agg_h @ lin_W + agg_e @ edge_Wpad + deg*(lin_b+edge_b))
// agg_h:[GN,64] (K=64, 16 steps), agg_e:[GN,16] padded (K=16, 4 steps).
// Full tiles only; 80 WMMAs per wave.
// ---------------------------------------------------------------------------
__global__ __launch_bounds__(256) void node_update_wmma(
    const float* __restrict__ aggh, const float* __restrict__ agge,
    const float* __restrict__ deg,  const float* __restrict__ W,
    const float* __restrict__ We,   const float* __restrict__ lb,
    const float* __restrict__ eb,   float* __restrict__ h)
{
    int wave = blockIdx.x * (blockDim.x >> 5) + (threadIdx.x >> 5);
    int lane = threadIdx.x & 31;
    int row0 = wave * 16;
    int m    = lane & 15;
    int half = lane >> 4;
    int khalf = half * 2;

    v2f aH[16], aE[4];
    const v2f* hr = (const v2f*)(aggh + (size_t)(row0 + m) * GH);
    const v2f* er = (const v2f*)(agge + (size_t)(row0 + m) * 16);
#pragma unroll
    for (int kk = 0; kk < 16; ++kk) aH[kk] = hr[kk * 2 + half];
#pragma unroll
    for (int kk = 0; kk < 4; ++kk)  aE[kk] = er[kk * 2 + half];

    // degree values for this lane's 8 output rows (hoisted out of j-loop)
    float degv[8];
#pragma unroll
    for (int rr = 0; rr < 8; ++rr) degv[rr] = deg[row0 + rr + 8 * half];

#pragma unroll
    for (int j = 0; j < 4; ++j) {
        v8f c = {};
        int col = j * 16 + m;
#pragma unroll
        for (int kk = 0; kk < 16; ++kk) {
            int k = kk * 4 + khalf;
            v2f bb;
            bb[0] = W[(size_t)k * GH + col];
            bb[1] = W[(size_t)(k + 1) * GH + col];
            c = __builtin_amdgcn_wmma_f32_16x16x4_f32(
                    false, aH[kk], false, bb, (short)0, c, false, false);
        }
#pragma unroll
        for (int kk = 0; kk < 4; ++kk) {
            int k = kk * 4 + khalf;
            v2f bb;
            bb[0] = We[(size_t)k * GH + col];
            bb[1] = We[(size_t)(k + 1) * GH + col];
            c = __builtin_amdgcn_wmma_f32_16x16x4_f32(
                    false, aE[kk], false, bb, (short)0, c, false, false);
        }
        float bias = lb[col] + eb[col];
        float* hp = h + (size_t)(row0 + 8 * half) * GH + col;
#pragma unroll
        for (int rr = 0; rr < 8; ++rr) {
            float v = c[rr] + degv[rr] * bias;
            hp[(size_t)rr * GH] = v > 0.0f ? v : 0.0f;
        }
    }
}

// ---------------------------------------------------------------------------
// Scores: scores[i] = relu(h[i]@att_W1 + b1) @ att_W2 + b2   (one wave/node)
// ---------------------------------------------------------------------------
__global__ __launch_bounds__(256) void scores_kernel(
    const float* __restrict__ h, const float* __restrict__ W1,
    const float* __restrict__ b1, const float* __restrict__ W2,
    const float* __restrict__ b2, float* __restrict__ scores)
{
    int wave = blockIdx.x * (blockDim.x >> 5) + (threadIdx.x >> 5);
    int lane = threadIdx.x & 31;
    if (wave >= GN) return;                      // wave-uniform
    const float* hrow = h + (size_t)wave * GH;
    float acc = 0.0f;
#pragma unroll
    for (int k = 0; k < GH; ++k) acc += hrow[k] * W1[k * 32 + lane];
    float s1 = acc + b1[lane];
    s1 = s1 > 0.0f ? s1 : 0.0f;
    float p = s1 * W2[lane];
#pragma unroll
    for (int off = 16; off > 0; off >>= 1) p += __shfl_xor(p, off, 32);
    if (lane == 0) scores[wave] = p + b2[0];
}

// ---------------------------------------------------------------------------
// Segment max via monotonic uint encoding + atomicMax
// ---------------------------------------------------------------------------
__device__ __forceinline__ unsigned enc_f32(float f) {
    unsigned u = __float_as_uint(f);
    return (u & 0x80000000u) ? ~u : (u | 0x80000000u);
}
__device__ __forceinline__ float dec_f32(unsigned u) {
    return (u & 0x80000000u) ? __uint_as_float(u & 0x7FFFFFFFu)
                             : __uint_as_float(~u);
}

__global__ void segmax_kernel(const float* __restrict__ scores,
                              const int* __restrict__ batch,
                              unsigned* __restrict__ menc)
{
    int i = blockIdx.x * blockDim.x + threadIdx.x;
    if (i >= GN) return;
    atomicMax(&menc[batch[i]], enc_f32(scores[i]));
}

__global__ void expdenom_kernel(const float* __restrict__ scores,
                                const int* __restrict__ batch,
                                const unsigned* __restrict__ menc,
                                float* __restrict__ ebuf,
                                float* __restrict__ denom)
{
    int i = blockIdx.x * blockDim.x + threadIdx.x;
    if (i >= GN) return;
    int g = batch[i];
    float ev = expf(scores[i] - dec_f32(menc[g]));
    ebuf[i] = ev;
    atomicAdd(&denom[g], ev);
}

// ---------------------------------------------------------------------------
// Attention pooling: gpool[g] += h[i] * (ebuf[i]/denom[g]), LDS-staged
// ---------------------------------------------------------------------------
__global__ __launch_bounds__(256) void attn_pool(
    const float* __restrict__ h, const float* __restrict__ ebuf,
    const float* __restrict__ denom, const int* __restrict__ batch,
    float* __restrict__ gpool)
{
    __shared__ float part[GG * GH];              // 16 KB
    for (int idx = threadIdx.x; idx < GG * GH; idx += blockDim.x) part[idx] = 0.0f;
    __syncthreads();

    int chunk = (GN + gridDim.x - 1) / gridDim.x;
    int start = blockIdx.x * chunk;
    int end   = start + chunk; if (end > GN) end = GN;
    int feat  = threadIdx.x & 63;
    int sub   = threadIdx.x >> 6;                // 0..3
    for (int i = start + sub; i < end; i += 4) {
        int g = batch[i];
        float a = ebuf[i] / denom[g];
        atomicAdd(&part[g * GH + feat], h[(size_t)i * GH + feat] * a);
    }
    __syncthreads();
    for (int idx = threadIdx.x; idx < GG * GH; idx += blockDim.x) {
        float v = part[idx];
        if (v != 0.0f) atomicAdd(&gpool[idx], v);
    }
}

// ---------------------------------------------------------------------------
// Decoder: out = relu(g@dec_W1 + b1) @ dec_W2 + b2   (64x64 -> 64x8, tiny,
// runs once on a single block -- keep loops rolled to avoid code bloat)
// ---------------------------------------------------------------------------
__global__ __launch_bounds__(64) void decoder_kernel(
    const float* __restrict__ g, const float* __restrict__ W1,
    const float* __restrict__ b1, const float* __restrict__ W2,
    const float* __restrict__ b2, float* __restrict__ out)
{
    int i = threadIdx.x;                         // graph row 0..63
    float t1[32];
#pragma unroll 1
    for (int j = 0; j < 32; ++j) {
        float a = b1[j];
#pragma unroll 4
        for (int k = 0; k < GH; ++k) a += g[i * GH + k] * W1[k * 32 + j];
        t1[j] = a > 0.0f ? a : 0.0f;
    }
#pragma unroll 1
    for (int o = 0; o < GDOUT; ++o) {
        float a = b2[o];
#pragma unroll 4
        for (int j = 0; j < 32; ++j) a += t1[j] * W2[j * GDOUT + o];
        out[i * GDOUT + o] = a;
    }
}

// ---------------------------------------------------------------------------
// Host-side orchestration
// ---------------------------------------------------------------------------
extern "C" void kernel_launch(void* const* d_in, const int* in_sizes, int n_in,
                              void* d_out, int out_size, void* d_ws, size_t ws_size,
                              hipStream_t stream) {
    const float* x      = (const float*)d_in[0];
    const int*   eidx   = (const int*)  d_in[1];
    const float* eattr  = (const float*)d_in[2];
    const int*   batch  = (const int*)  d_in[3];
    const float* enc_W  = (const float*)d_in[4];
    const float* enc_b  = (const float*)d_in[5];
    const float* lin_W  = (const float*)d_in[6];
    const float* lin_b  = (const float*)d_in[7];
    const float* edge_W = (const float*)d_in[8];
    const float* edge_b = (const float*)d_in[9];
    const float* att_W1 = (const float*)d_in[10];
    const float* att_b1 = (const float*)d_in[11];
    const float* att_W2 = (const float*)d_in[12];
    const float* att_b2 = (const float*)d_in[13];
    const float* dec_W1 = (const float*)d_in[14];
    const float* dec_b1 = (const float*)d_in[15];
    const float* dec_W2 = (const float*)d_in[16];
    const float* dec_b2 = (const float*)d_in[17];

    // Workspace layout (floats)
    float* ws     = (float*)d_ws;
    float* h      = ws;                                  // GN*64
    float* aggh   = h     + (size_t)GN * GH;             // GN*64
    float* agge   = aggh  + (size_t)GN * GH;             // GN*16
    float* deg    = agge  + (size_t)GN * 16;             // GN
    float* scores = deg   + GN;                          // GN
    float* ebuf   = scores + GN;                         // GN
    float* gpool  = ebuf  + GN;                          // 64*64
    float* denom  = gpool + GG * GH;                     // 64
    unsigned* menc = (unsigned*)(denom + GG);            // 64
    float* Wpad   = (float*)(menc + GG);                 // 2*16*64

    const int* srcI = eidx;
    const int* dstI = eidx + GE;

    // Pad edge weights once
    pad_edge_w<<<(2 * 16 * 64 + 255) / 256, 256, 0, stream>>>(edge_W, Wpad);

    // Encoder: exactly GN/16 = 6250 waves, 8 waves/block -> pad grid; the
    // tile count divides evenly so every launched wave that passes the
    // uniform bound check does a full tile.  6250 waves / 8 = 781.25 -> use
    // 625 blocks of 10 tiles? Keep it simple: 6250 waves = 782 blocks with a
    // uniform guard would reintroduce a branch; instead use 250 blocks x 25
    // waves? Max block = 1024 threads = 32 waves. 6250 = 25 * 250: launch
    // 250 blocks of 800 threads (25 waves), exact cover, zero guards.
    encoder_wmma<<<250, 800, 0, stream>>>(x, enc_W, enc_b, h);

    // Message-passing layers
    for (int l = 0; l < 2; ++l) {
        zero_f32<<<2048, 256, 0, stream>>>(aggh, (size_t)GN * (GH + 16 + 1));
        size_t sthreads = (size_t)GE * 64;
        int blksE = (int)((sthreads + 255) / 256);
        scatter_edges<<<blksE, 256, 0, stream>>>(h, srcI, dstI, eattr,
                                                 aggh, agge, deg);
        node_update_wmma<<<250, 800, 0, stream>>>(
            aggh, agge, deg,
            lin_W + (size_t)l * GH * GH,
            Wpad  + (size_t)l * 16 * GH,
            lin_b + (size_t)l * GH,
            edge_b + (size_t)l * GH,
            h);
    }

    // Attention pooling
    int blocksS = (GN + 7) / 8;                          // wave per node
    scores_kernel<<<blocksS, 256, 0, stream>>>(h, att_W1, att_b1,
                                               att_W2, att_b2, scores);
    zero_f32<<<32, 256, 0, stream>>>(gpool, (size_t)(GG * GH + GG + GG));
    segmax_kernel<<<(GN + 255) / 256, 256, 0, stream>>>(scores, batch, menc);
    expdenom_kernel<<<(GN + 255) / 256, 256, 0, stream>>>(scores, batch, menc,
                                                          ebuf, denom);
    attn_pool<<<128, 256, 0, stream>>>(h, ebuf, denom, batch, gpool);

    // Decoder
    decoder_kernel<<<1, 64, 0, stream>>>(gpool, dec_W1, dec_b1,
                                         dec_W2, dec_b2, (float*)d_out);
}